// LSTM_45664092291561
// MI455X (gfx1250) — compile-verified
//
#include <hip/hip_runtime.h>

#ifndef __has_builtin
#define __has_builtin(x) 0
#endif

#define AS1 __attribute__((address_space(1)))
#define AS3 __attribute__((address_space(3)))

typedef int v2i __attribute__((vector_size(8)));    // b64 payload type for async builtins
typedef int v4i __attribute__((vector_size(16)));   // b128 payload type for async builtins
typedef unsigned int u32x4 __attribute__((ext_vector_type(4)));
typedef int i32x8 __attribute__((ext_vector_type(8)));
typedef int i32x4 __attribute__((ext_vector_type(4)));

// Problem constants: x[B,T,1], w_ih[8,1], w_hh[8,2], out = [B,T,2] ++ [B,2] ++ [B,2]
static constexpr int Bsz   = 8192;
static constexpr int Tn    = 1024;
static constexpr int Hn    = 2;
static constexpr int NB    = 32;    // batch rows per block == blockDim.x == 1 wave32
static constexpr int K     = 64;    // timesteps per tile
static constexpr int TILES = Tn / K;            // 16
static constexpr int XST   = K + 2;             // 66: x-tile LDS row stride, conflict-free reads
static constexpr int OST   = 2 * K + 4;         // 132: out-tile LDS row stride, 16B-aligned rows
static constexpr int SPT   = (K * Hn) / 4;      // 32 b128 store chunks per output row

#if __has_builtin(__builtin_amdgcn_tensor_load_to_lds)
#define HAVE_TDM 1
#else
#define HAVE_TDM 0
#endif

#define ASYNC_WAIT(n) __builtin_amdgcn_s_wait_asynccnt(n)

// ---------- async LDS -> global stores (ASYNCcnt) ----------
__device__ __forceinline__ void async_store_b64(float* g, const float* l) {
#if __has_builtin(__builtin_amdgcn_global_store_async_from_lds_b64)
  __builtin_amdgcn_global_store_async_from_lds_b64(
      (AS1 v2i*)(AS1 void*)g, (AS3 v2i*)(AS3 void*)l, 0, 0);
#else
  unsigned loff = (unsigned)(unsigned long long)(AS3 void*)l;
  asm volatile("global_store_async_from_lds_b64 %0, %1, off" :: "v"(g), "v"(loff) : "memory");
#endif
}

__device__ __forceinline__ void async_store_b128(float* g, const float* l) {
#if __has_builtin(__builtin_amdgcn_global_store_async_from_lds_b128)
  __builtin_amdgcn_global_store_async_from_lds_b128(
      (AS1 v4i*)(AS1 void*)g, (AS3 v4i*)(AS3 void*)l, 0, 0);
#else
  async_store_b64(g, l);
  async_store_b64(g + 2, l + 2);
#endif
}

// ---------- x-tile staging: TDM tensor load (TENSORcnt) with async fallback ----------
__device__ __forceinline__ void async_load_b64(const float* g, float* l) {
#if __has_builtin(__builtin_amdgcn_global_load_async_to_lds_b64)
  __builtin_amdgcn_global_load_async_to_lds_b64(
      (AS1 v2i*)(AS1 void*)g, (AS3 v2i*)(AS3 void*)l, 0, 0);
#else
  unsigned loff = (unsigned)(unsigned long long)(AS3 void*)l;
  asm volatile("global_load_async_to_lds_b64 %0, %1, off" :: "v"(loff), "v"(g) : "memory");
#endif
}

// Load a 2D tile x[b0 .. b0+31][t0 .. t0+K-1] (row stride Tn) into LDS rows of
// stride XST floats (TDM pad: 64 dwords data + 2 dwords pad per row).
__device__ __forceinline__ void load_x_tile(const float* gtile, float* ltile, int tid) {
#if HAVE_TDM
  (void)tid;
  unsigned lds_off = (unsigned)(unsigned long long)(AS3 void*)ltile;
  unsigned long long ga = (unsigned long long)gtile;
  u32x4 g0;
  g0.x = 1u;                                        // count=1, user descriptor
  g0.y = lds_off;                                   // lds_addr (bytes)
  g0.z = (unsigned)(ga & 0xFFFFFFFFu);              // global_addr[31:0]
  g0.w = (unsigned)((ga >> 32) & 0x01FFFFFFu) | 0x80000000u;  // addr[56:32] | type=2
  i32x8 g1;
  g1[0] = (2 << 16)        // data_size = 4 bytes
        | (1 << 20)        // pad_enable
        | (5 << 22)        // pad_interval: 64 dwords
        | (1 << 25);       // pad_amount: 2 dwords  -> LDS row stride = 66 dwords
  g1[1] = (int)0xFFFF0000u;            // tensor_dim0 = 0x7FFFFFFF (lo16 in [31:16]); atomic addr = 0
  g1[2] = (int)0xFFFF7FFFu;            // tensor_dim0 hi16 | tensor_dim1 lo16 (dim1 = 0x7FFFFFFF)
  g1[3] = (K << 16) | 0x7FFF;          // tensor_dim1 hi16 | tile_dim0 = K
  g1[4] = NB;                          // tile_dim1 = 32 rows, tile_dim2 = 0
  g1[5] = Tn;                          // tensor_dim0_stride = 1024 (lo32)
  g1[6] = 0;                           // stride hi16, dim1_stride lo16
  g1[7] = 0;
  i32x4 z4 = {0, 0, 0, 0};
  i32x8 z8 = {0, 0, 0, 0, 0, 0, 0, 0};
  // 6-arg variant (this toolchain): (g0, g1, g2, g3, extra, cpol)
  __builtin_amdgcn_tensor_load_to_lds(g0, g1, z4, z4, z8, 0);
#else
  // Fallback: per-lane async b64 loads (lane tid takes column pair 2*tid of row it).
#pragma unroll
  for (int it = 0; it < NB; ++it)
    async_load_b64(gtile + (size_t)it * Tn + 2 * tid, ltile + it * XST + 2 * tid);
#endif
}

// Wait helpers: with TDM, x-loads use TENSORcnt (prefetch depth 1). Without it,
// loads share ASYNCcnt with stores -> conservative full drains keep correctness.
__device__ __forceinline__ void wait_xload_allow1() {
#if HAVE_TDM
  __builtin_amdgcn_s_wait_tensorcnt(1);
#else
  ASYNC_WAIT(0);
#endif
}
__device__ __forceinline__ void wait_xload_all() {
#if HAVE_TDM
  __builtin_amdgcn_s_wait_tensorcnt(0);
#else
  ASYNC_WAIT(0);
#endif
}
__device__ __forceinline__ void wait_store_recycle() {
#if HAVE_TDM
  ASYNC_WAIT(SPT);   // all but newest tile's 32 stores drained (stores complete in order)
#else
  ASYNC_WAIT(0);
#endif
}

__device__ __forceinline__ float sigm(float x)  { return 1.0f / (1.0f + __expf(-x)); }
__device__ __forceinline__ float ftanh(float x) { float e = __expf(2.0f * x); return (e - 1.0f) / (e + 1.0f); }

__global__ __launch_bounds__(NB)
void LSTM_45664092291561_kernel(const float* __restrict__ x,
                                const float* __restrict__ w_ih,
                                const float* __restrict__ w_hh,
                                float* __restrict__ out) {
  __shared__ __align__(16) float ldsx[2][NB * XST];  // double-buffered x tiles
  __shared__ __align__(16) float ldso[2][NB * OST];  // double-buffered h tiles
  const int tid = threadIdx.x;          // 0..31
  const int b0  = blockIdx.x * NB;
  const int b   = b0 + tid;

  // Uniform weights -> scalar loads. Gate order i,f,g,o; each gate has H=2 rows.
  float wi[8], a0[8], a1[8];
#pragma unroll
  for (int gi = 0; gi < 8; ++gi) {
    wi[gi] = w_ih[gi];
    a0[gi] = w_hh[2 * gi + 0];
    a1[gi] = w_hh[2 * gi + 1];
  }

  float h0 = 0.f, h1 = 0.f, c0 = 0.f, c1 = 0.f;

  // Prefetch tile 0.
  load_x_tile(x + (size_t)b0 * Tn, &ldsx[0][0], tid);

  for (int n = 0; n < TILES; ++n) {
    const int t0 = n * K;

    // Prefetch next tile into the other x buffer, then wait only for tile n.
    if (n + 1 < TILES) {
      load_x_tile(x + (size_t)b0 * Tn + (n + 1) * K, &ldsx[(n + 1) & 1][0], tid);
      wait_xload_allow1();
    } else {
      wait_xload_all();
    }
    // Recycle the output buffer: stores from tile n-2 must be fully drained.
    wait_store_recycle();

    const float* xrow = &ldsx[n & 1][tid * XST];
    float*       orow = &ldso[n & 1][tid * OST];

    // Recurrence: K steps in registers; conflict-free LDS reads, 2-way-conflict writes.
#pragma unroll 4
    for (int j = 0; j < K; ++j) {
      float xt = xrow[j];
      float p0 = fmaf(h1, a1[0], fmaf(h0, a0[0], xt * wi[0]));
      float p1 = fmaf(h1, a1[1], fmaf(h0, a0[1], xt * wi[1]));
      float p2 = fmaf(h1, a1[2], fmaf(h0, a0[2], xt * wi[2]));
      float p3 = fmaf(h1, a1[3], fmaf(h0, a0[3], xt * wi[3]));
      float p4 = fmaf(h1, a1[4], fmaf(h0, a0[4], xt * wi[4]));
      float p5 = fmaf(h1, a1[5], fmaf(h0, a0[5], xt * wi[5]));
      float p6 = fmaf(h1, a1[6], fmaf(h0, a0[6], xt * wi[6]));
      float p7 = fmaf(h1, a1[7], fmaf(h0, a0[7], xt * wi[7]));
      float i0 = sigm(p0),  i1 = sigm(p1);
      float f0 = sigm(p2),  f1 = sigm(p3);
      float g0 = ftanh(p4), g1 = ftanh(p5);
      float o0 = sigm(p6),  o1 = sigm(p7);
      c0 = fmaf(f0, c0, i0 * g0);
      c1 = fmaf(f1, c1, i1 * g1);
      h0 = o0 * ftanh(c0);
      h1 = o1 * ftanh(c1);
      *reinterpret_cast<float2*>(&orow[2 * j]) = make_float2(h0, h1);
    }

    // Drain tile n: 32 b128 lane-ops, each wavefront op = 512B contiguous in global.
#pragma unroll
    for (int it = 0; it < SPT; ++it) {
      float* g = out + ((size_t)(b0 + it) * Tn + t0) * Hn + 4 * tid;
      async_store_b128(g, &ldso[n & 1][it * OST + 4 * tid]);
    }
  }

  // hT / cT tails (coalesced float2 per lane).
  float* ht = out + (size_t)Bsz * Tn * Hn;
  float* ct = ht + (size_t)Bsz * Hn;
  *reinterpret_cast<float2*>(&ht[2 * b]) = make_float2(h0, h1);
  *reinterpret_cast<float2*>(&ct[2 * b]) = make_float2(c0, c1);

  ASYNC_WAIT(0);
  wait_xload_all();
}

extern "C" void kernel_launch(void* const* d_in, const int* in_sizes, int n_in,
                              void* d_out, int out_size, void* d_ws, size_t ws_size,
                              hipStream_t stream) {
  (void)in_sizes; (void)n_in; (void)out_size; (void)d_ws; (void)ws_size;
  const float* x    = (const float*)d_in[0];
  const float* w_ih = (const float*)d_in[1];
  const float* w_hh = (const float*)d_in[2];
  float* out = (float*)d_out;
  dim3 grid(Bsz / NB), block(NB);
  hipLaunchKernelGGL(LSTM_45664092291561_kernel, grid, block, 0, stream, x, w_ih, w_hh, out);
}